// EarModel_42082089566398
// MI455X (gfx1250) — compile-verified
//
#include <hip/hip_runtime.h>
#include <math.h>

#ifndef M_PI
#define M_PI 3.14159265358979323846
#endif

#define FS_I   24000
#define T_LEN  24000
#define NCHAN  32
#define BATCH  8
#define TILE   256
#define NTILE  94            // ceil(24000/256)
#define SMALLF 1e-30f

typedef float v2f __attribute__((ext_vector_type(2)));
typedef float v8f __attribute__((ext_vector_type(8)));

// ---------------- ERB center frequencies (ascending), double precision ----------------
__device__ __forceinline__ double cf_of(int i) {
  const double c = 9.26449 * 24.7;
  if (i == 31) return 8000.0;
  double j = (double)(31 - i);
  return -c + exp(j * (log(80.0 + c) - log(8000.0 + c)) / 31.0) * (8000.0 + c);
}

// ---------------- demodulation tables: cos(n w), -sin(n w), f64 phase ----------------
__global__ void k_tables(float* __restrict__ cosT, float* __restrict__ sinT) {
  int idx = blockIdx.x * blockDim.x + threadIdx.x;
  if (idx >= NCHAN * T_LEN) return;
  int ch = idx / T_LEN, t = idx % T_LEN;
  double ph = (2.0 * M_PI / 24000.0) * cf_of(ch) * (double)t;
  cosT[idx] = (float)cos(ph);
  sinT[idx] = (float)(-sin(ph));
}

// ---------------- loss_parameters for both ears + bw_max ----------------
__global__ void k_params(const float* __restrict__ hl,
                         float* attn_ohc, float* bw_min, float* low_knee,
                         float* crv, float* attn_ihc, float* bw_max) {
  int tid = threadIdx.x;               // 256 = BATCH*NCHAN
  int b = tid / NCHAN, ch = tid % NCHAN;
  float cfc = (float)cf_of(ch);
  float fv[8];
  fv[0] = (float)cf_of(0);
  fv[1] = 250.f; fv[2] = 500.f; fv[3] = 1000.f;
  fv[4] = 2000.f; fv[5] = 4000.f; fv[6] = 6000.f;
  fv[7] = (float)cf_of(31);
  float cr    = 1.25f + 2.25f * (float)ch / 31.0f;
  float theo  = 1.25f * 70.0f * (1.0f - 1.0f / cr);
  float upamp = 30.0f + 70.0f / cr;
  for (int e = 0; e < 2; ++e) {
    float val[8];
    float h0 = (e == 0) ? 0.f : hl[b * 6 + 0];
    val[0] = h0;
    for (int i = 0; i < 6; ++i) val[i + 1] = (e == 0) ? 0.f : hl[b * 6 + i];
    val[7] = val[6];
    float loss;
    if (cfc <= fv[0]) loss = val[0];
    else if (cfc >= fv[7]) loss = val[7];
    else {
      loss = val[7];
      for (int i = 0; i < 7; ++i)
        if (cfc < fv[i + 1]) {
          float tt = (cfc - fv[i]) / (fv[i + 1] - fv[i]);
          loss = val[i] + tt * (val[i + 1] - val[i]);
          break;
        }
    }
    loss = fmaxf(loss, 0.f);
    float aohc = (loss < theo) ? 0.8f * loss : 0.8f * theo;
    float aihc = (loss < theo) ? 0.2f * loss : 0.2f * theo + (loss - theo);
    float q = aohc / 50.f;
    float bmin = 1.f + q + 2.f * q * q * q;
    float lk = aohc + 30.f;
    float cro = (100.f - lk) / (upamp + aohc - lk);
    int o = (e * BATCH + b) * NCHAN + ch;
    attn_ohc[o] = aohc; bw_min[o] = bmin; low_knee[o] = lk;
    crv[o] = cro; attn_ihc[o] = aihc;
  }
  if (b == 0) {  // bw_max from 100 dB loss (theoretical max OHC loss path)
    float aohc = 0.8f * theo;  // 100 > theo always
    float q = aohc / 50.f;
    bw_max[ch] = 1.f + q + 2.f * q * q * q;
  }
}

// ---------------- middle ear: LP5k (1st) then HP350 (2nd), 16 serial chains ----------------
__global__ void k_mid(const float* __restrict__ ref, const float* __restrict__ proc,
                      float* __restrict__ mid) {
  int id = threadIdx.x;
  if (id >= 16) return;
  int e = id / BATCH, b = id % BATCH;
  const float* x = (e == 0 ? ref : proc) + b * T_LEN;
  float* y = mid + (size_t)(e * BATCH + b) * T_LEN;
  double cl = tan(M_PI * 5000.0 / 24000.0);
  float lb = (float)(cl / (1.0 + cl));
  float la = (float)((cl - 1.0) / (1.0 + cl));
  double cc = tan(M_PI * 350.0 / 24000.0);
  double d = 1.0 + sqrt(2.0) * cc + cc * cc;
  float hb0 = (float)(1.0 / d), hb1 = (float)(-2.0 / d), hb2 = hb0;
  float ha1 = (float)(2.0 * (cc * cc - 1.0) / d);
  float ha2 = (float)((1.0 - sqrt(2.0) * cc + cc * cc) / d);
  float z = 0.f, w0 = 0.f, w1 = 0.f;
  for (int t = 0; t < T_LEN; ++t) {
    float xt = x[t];
    float y1 = lb * xt + z;
    z = lb * xt - la * y1;
    float y2 = hb0 * y1 + w0;
    w0 = hb1 * y1 - ha1 * y2 + w1;
    w1 = hb2 * y1 - ha2 * y2;
    y[t] = y2;
  }
}

// ---------------- WMMA blocked-scan machinery ----------------
// Tile = 256 samples = 16x16 matrix, X[t][c] = x[c*16+t].
// First-order recurrence y = a*y_prev + x  ==>  Y = L*X (+ carry fix), L[t][j]=a^(t-j), j<=t.

__device__ __forceinline__ float selp(const float p[16], int idx) {
  float r = p[0];
#pragma unroll
  for (int i = 1; i < 16; ++i) r = (idx == i) ? p[i] : r;
  return r;
}

struct FO {
  v2f A[4];        // L in 16x4 A-chunks
  float apow1[8];  // a^(t+1) for this lane's 8 rows
  float a16;       // a^16
};

__device__ __forceinline__ void build_fo(float a, FO& f, int lane) {
  float p[16];
  p[0] = 1.f;
#pragma unroll
  for (int i = 1; i < 16; ++i) p[i] = p[i - 1] * a;
  f.a16 = p[15] * a;
  int m = lane & 15;
  int jb = (lane >= 16) ? 2 : 0;  // A layout: lo lanes K0,K1 ; hi lanes K2,K3
#pragma unroll
  for (int k = 0; k < 4; ++k) {
    int j0 = 4 * k + jb, j1 = j0 + 1;
    v2f v;
    v.x = (j0 <= m) ? selp(p, m - j0) : 0.f;
    v.y = (j1 <= m) ? selp(p, m - j1) : 0.f;
    f.A[k] = v;
  }
  int tb = (lane >= 16) ? 8 : 0;
#pragma unroll
  for (int r = 0; r < 8; ++r) f.apow1[r] = selp(p, r + tb) * a;
}

// One first-order scan stage on a tile: B-layout in, D-layout out, carry state s.
__device__ __forceinline__ void fo_scan(const FO& f, const v2f B[4], float& s,
                                        float out[8], int lane) {
  v8f acc = {0.f, 0.f, 0.f, 0.f, 0.f, 0.f, 0.f, 0.f};
  acc = __builtin_amdgcn_wmma_f32_16x16x4_f32(false, f.A[0], false, B[0], (short)0, acc, false, false);
  acc = __builtin_amdgcn_wmma_f32_16x16x4_f32(false, f.A[1], false, B[1], (short)0, acc, false, false);
  acc = __builtin_amdgcn_wmma_f32_16x16x4_f32(false, f.A[2], false, B[2], (short)0, acc, false, false);
  acc = __builtin_amdgcn_wmma_f32_16x16x4_f32(false, f.A[3], false, B[3], (short)0, acc, false, false);
  float y15 = acc[7];          // row t=15 lives in lanes 16..31 of reg 7
  int myc = lane & 15;
  float s_run = s, s_mine = s; // s_c = state entering column c
#pragma unroll
  for (int j = 0; j < 16; ++j) {
    float yj = __shfl(y15, 16 + j, 32);
    if (j == myc) s_mine = s_run;
    s_run = yj + f.a16 * s_run;      // s_{j+1} = Y15[j] + a^16 * s_j
  }
  s = s_run;
#pragma unroll
  for (int r = 0; r < 8; ++r) out[r] = acc[r] + f.apow1[r] * s_mine;
}

// D-layout -> LDS (stride-17 to dodge bank conflicts) -> B-layout. Same-wave DS is in-order.
__device__ __forceinline__ void d_to_B(const float d[8], volatile float* lds,
                                       v2f B[4], int lane) {
  int c = lane & 15;
  int tb = (lane >= 16) ? 8 : 0;
#pragma unroll
  for (int r = 0; r < 8; ++r) lds[c * 17 + r + tb] = d[r];
  int toff = (lane >= 16) ? 2 : 0;  // B layout: lo lanes K={0,1}, hi lanes K={2,3}
#pragma unroll
  for (int k = 0; k < 4; ++k) {
    v2f v;
    v.x = lds[c * 17 + 4 * k + toff];
    v.y = lds[c * 17 + 4 * k + toff + 1];
    B[k] = v;
  }
}

__device__ __forceinline__ float demod_at(const float* m, const float* trig, int g) {
  return (g >= 0 && g < T_LEN) ? m[g] * trig[g] : 0.f;
}

// Build B-layout input for the cascade: FIR(1, 4a, 4a^2) of demodulated signal.
__device__ __forceinline__ void build_B_input(const float* m, const float* trig,
                                              int base, float fa1, float fa2,
                                              v2f B[4], int lane) {
  int c = lane & 15;
  int toff = (lane >= 16) ? 2 : 0;
#pragma unroll
  for (int k = 0; k < 4; ++k) {
    v2f v;
#pragma unroll
    for (int u = 0; u < 2; ++u) {
      int t = 4 * k + toff + u;
      int g = base + c * 16 + t;
      float f = demod_at(m, trig, g) + fa1 * demod_at(m, trig, g - 1)
              + fa2 * demod_at(m, trig, g - 2);
      if (u == 0) v.x = f; else v.y = f;
    }
    B[k] = v;
  }
}

// 4-stage cascade (= denominator (1-az)^4) on one tile; result in D layout.
__device__ __forceinline__ void gt_tile(const FO& f, float fa1, float fa2,
                                        const float* m, const float* trig, int base,
                                        float st[4], volatile float* lds,
                                        float u[8], int lane) {
  v2f B[4];
  build_B_input(m, trig, base, fa1, fa2, B, lane);
  float tmp[8];
#pragma unroll
  for (int sgi = 0; sgi < 4; ++sgi) {
    fo_scan(f, B, st[sgi], tmp, lane);
    if (sgi < 3) d_to_B(tmp, lds, B, lane);
  }
#pragma unroll
  for (int r = 0; r < 8; ++r) u[r] = tmp[r];
}

__device__ __forceinline__ void gt_coeffs(float bw, float cfc, float& a, float& fa1,
                                          float& fa2, float& gg) {
  float tpt = 2.0f * (float)M_PI / 24000.0f;
  float tptbw = tpt * bw * 1.019f * (24.7f + cfc / 9.26449f);
  a = expf(-tptbw);
  float a1 = 4.f * a, a2 = -6.f * a * a, a3 = 4.f * a * a * a;
  float a4 = -(a * a * a * a), a5 = 4.f * a * a;
  fa1 = a1; fa2 = a5;
  gg = 2.f * (1.f - a1 - a2 - a3 - a4) / (1.f + a1 + a5);
}

// ---------------- main: one wave per (ear,batch,chan) chain ----------------
__global__ void __launch_bounds__(256) k_main(
    const float* __restrict__ mid, const float* __restrict__ cosT,
    const float* __restrict__ sinT, const float* __restrict__ attn_ohc_a,
    const float* __restrict__ bw_min_a, const float* __restrict__ low_knee_a,
    const float* __restrict__ cr_a, const float* __restrict__ attn_ihc_a,
    const float* __restrict__ bw_max_a, const int* __restrict__ lvl,
    float* __restrict__ out) {
  __shared__ float ldsbuf[8][17 * 16];
  int lane = threadIdx.x & 31;
  int wave = threadIdx.x >> 5;
  int id = blockIdx.x * 8 + wave;   // 0..511
  int e = id >> 8, b = (id >> 5) & 7, ch = id & 31;
  volatile float* lds = ldsbuf[wave];
  float lv1 = (float)(*lvl);
  float cfc = (float)cf_of(ch);
  int pidx = (e * BATCH + b) * NCHAN + ch;
  float aohc = attn_ohc_a[pidx], bmin = bw_min_a[pidx], lk = low_knee_a[pidx];
  float crr = cr_a[pidx], aihc = attn_ihc_a[pidx], bmax = bw_max_a[ch];
  const float* midp = mid + (size_t)(e * BATCH + b) * T_LEN;
  const float* cosp = cosT + (size_t)ch * T_LEN;
  const float* sinp = sinT + (size_t)ch * T_LEN;

  // ---- control gammatone (bw_max): RMS of control envelope ----
  float ac, fa1c, fa2c, ggc;
  gt_coeffs(bmax, cfc, ac, fa1c, fa2c, ggc);
  FO foc; build_fo(ac, foc, lane);
  int c = lane & 15, tb = (lane >= 16) ? 8 : 0, toff = (lane >= 16) ? 2 : 0;
  float sumsq = 0.f;
  {
    float str[4] = {0, 0, 0, 0}, sti[4] = {0, 0, 0, 0};
    for (int tile = 0; tile < NTILE; ++tile) {
      int base = tile * TILE;
      float ur[8], ui[8];
      gt_tile(foc, fa1c, fa2c, midp, cosp, base, str, lds, ur, lane);
      gt_tile(foc, fa1c, fa2c, midp, sinp, base, sti, lds, ui, lane);
#pragma unroll
      for (int r = 0; r < 8; ++r) {
        int g = base + c * 16 + r + tb;
        float env = ggc * sqrtf(ur[r] * ur[r] + ui[r] * ui[r]);
        if (g < T_LEN) sumsq += env * env;
      }
    }
  }
#pragma unroll
  for (int o = 16; o > 0; o >>= 1) sumsq += __shfl_xor(sumsq, o, 32);
  float cdb = 20.f * log10f(sqrtf(sumsq / (float)T_LEN) + SMALLF) + lv1;
  float bw = (cdb < 50.f) ? bmin
           : ((cdb > 100.f) ? bmax : bmin + ((cdb - 50.f) / 50.f) * (bmax - bmin));

  // ---- signal pass: signal gammatone + control recompute + compression ----
  float as, fa1s, fa2s, ggs;
  gt_coeffs(bw, cfc, as, fa1s, fa2s, ggs);
  FO fos; build_fo(as, fos, lane);
  double clp = tan(M_PI * 800.0 / 24000.0);
  float blp = (float)(clp / (1.0 + clp));
  float alp = (float)(-((clp - 1.0) / (1.0 + clp)));  // y = alp*y_prev + blp*(x+x_prev)
  FO folp; build_fo(alp, folp, lane);
  float onem = 1.f - 1.f / crr;
  float str[4] = {0, 0, 0, 0}, sti[4] = {0, 0, 0, 0};
  float scr[4] = {0, 0, 0, 0}, sci[4] = {0, 0, 0, 0};
  float slp = 0.f, gprev = 0.f;
  float* outp = out + (size_t)((e * BATCH + b) * NCHAN + ch) * T_LEN;
  for (int tile = 0; tile < NTILE; ++tile) {
    int base = tile * TILE;
    float ur[8], ui[8], vr[8], vi[8];
    gt_tile(fos, fa1s, fa2s, midp, cosp, base, str, lds, ur, lane);
    gt_tile(fos, fa1s, fa2s, midp, sinp, base, sti, lds, ui, lane);
    gt_tile(foc, fa1c, fa2c, midp, cosp, base, scr, lds, vr, lane);
    gt_tile(foc, fa1c, fa2c, midp, sinp, base, sci, lds, vi, lane);
    float graw[8];
#pragma unroll
    for (int r = 0; r < 8; ++r) {
      float ctl = ggc * sqrtf(vr[r] * vr[r] + vi[r] * vi[r]);
      float logenv = lv1 + 20.f * log10f(fmaxf(ctl, SMALLF));
      logenv = fminf(fmaxf(logenv, lk), 100.f);
      graw[r] = exp10f((-aohc - (logenv - lk) * onem) * 0.05f);
    }
    // gain smoothing LP800 via the same WMMA scan: stage input FIR = blp*(g[t]+g[t-1])
#pragma unroll
    for (int r = 0; r < 8; ++r) lds[c * 17 + r + tb] = graw[r];
    v2f B[4];
#pragma unroll
    for (int k = 0; k < 4; ++k) {
      v2f v;
#pragma unroll
      for (int u = 0; u < 2; ++u) {
        int t = 4 * k + toff + u;
        float g0 = lds[c * 17 + t];
        float gm1;
        if (t > 0) gm1 = lds[c * 17 + t - 1];
        else if (c > 0) gm1 = lds[(c - 1) * 17 + 15];
        else gm1 = gprev;
        float f = blp * (g0 + gm1);
        if (u == 0) v.x = f; else v.y = f;
      }
      B[k] = v;
    }
    gprev = lds[15 * 17 + 15];  // last raw gain of this tile -> x[t-1] for next tile
    float gs[8];
    fo_scan(folp, B, slp, gs, lane);
#pragma unroll
    for (int r = 0; r < 8; ++r) {
      int g = base + c * 16 + r + tb;
      float envs = ggs * sqrtf(ur[r] * ur[r] + ui[r] * ui[r]);
      float ec = gs[r] * envs;
      float o = fmaxf(lv1 - aihc + 20.f * log10f(ec + SMALLF), 0.f);
      if (g < T_LEN) outp[g] = o;
    }
  }
}

extern "C" void kernel_launch(void* const* d_in, const int* in_sizes, int n_in,
                              void* d_out, int out_size, void* d_ws, size_t ws_size,
                              hipStream_t stream) {
  (void)in_sizes; (void)n_in; (void)out_size; (void)ws_size;
  const float* ref  = (const float*)d_in[0];
  const float* proc = (const float*)d_in[1];
  const float* hl   = (const float*)d_in[2];
  const int*   lvl  = (const int*)d_in[3];
  float* ws = (float*)d_ws;
  float* cosT = ws;                               // 32*24000
  float* sinT = cosT + NCHAN * T_LEN;             // 32*24000
  float* mid  = sinT + NCHAN * T_LEN;             // 2*8*24000
  float* attn_ohc = mid + 2 * BATCH * T_LEN;      // 512 each below
  float* bw_min   = attn_ohc + 512;
  float* low_knee = bw_min + 512;
  float* crv      = low_knee + 512;
  float* attn_ihc = crv + 512;
  float* bw_max   = attn_ihc + 512;               // 32

  k_tables<<<(NCHAN * T_LEN + 255) / 256, 256, 0, stream>>>(cosT, sinT);
  k_params<<<1, 256, 0, stream>>>(hl, attn_ohc, bw_min, low_knee, crv, attn_ihc, bw_max);
  k_mid<<<1, 16, 0, stream>>>(ref, proc, mid);
  k_main<<<64, 256, 0, stream>>>(mid, cosT, sinT, attn_ohc, bw_min, low_knee,
                                 crv, attn_ihc, bw_max, lvl, (float*)d_out);
}